// Objective_Func_63153199121079
// MI455X (gfx1250) — compile-verified
//
#include <hip/hip_runtime.h>
#include <hip/hip_bf16.h>

// Problem constants (match reference)
#define NB   64      // N_BLOCKS
#define NL   1024    // N_LINKS
#define TT   1000    // time steps (T_MAX/DT)
#define DTF  0.01f
#define LN2F 0.69314718056f

typedef float v2f __attribute__((ext_vector_type(2)));
typedef float v8f __attribute__((ext_vector_type(8)));

// ---------------------------------------------------------------------------
// Kernel A: sinr[n,i] via WMMA f32 16x16x4 GEMM  S[n,i] = sum_j P[n,j]*G[i,j]
// One wave32 computes one 16x16 (n x i) tile. 4 n-tiles * 64 i-tiles = 256.
// A (16x4 f32): lanes 0-15 hold M=0..15 K={0,1}; lanes 16-31 K={2,3}.
// B (4x16 f32): lanes 0-15 hold N=0..15 K={0,1}; lanes 16-31 K={2,3}.
// C/D (16x16 f32): lane c<16 -> N=c, M=vgpr; lane c>=16 -> N=c-16, M=vgpr+8.
// ---------------------------------------------------------------------------
__global__ __launch_bounds__(256) void sinr_wmma_kernel(
    const float* __restrict__ powers,    // [NB, NL]
    const float* __restrict__ pathloss,  // [NL, NL]
    float* __restrict__ sinr)            // [NB, NL]
{
    const int waveId = blockIdx.x * (blockDim.x >> 5) + (threadIdx.x >> 5); // 0..255
    const int n_t  = waveId >> 6;      // 0..3
    const int i_t  = waveId & 63;      // 0..63
    const int lane = threadIdx.x & 31;
    const int row  = lane & 15;
    const int half = lane >> 4;
    const int nb   = n_t * 16;
    const int ib   = i_t * 16;
    const int off  = half * 2;         // K sub-offset for this half-wave

    const float* __restrict__ Arow = powers   + (nb + row) * NL;  // A[m=row, k]
    const float* __restrict__ Brow = pathloss + (ib + row) * NL;  // B[k, col=row] = G[i, j=k]

    v8f acc = {0.f, 0.f, 0.f, 0.f, 0.f, 0.f, 0.f, 0.f};

    for (int jb = 0; jb < NL; jb += 4) {
        v2f a = *(const v2f*)(Arow + jb + off);   // 8B aligned: jb%4==0, off even
        v2f b = *(const v2f*)(Brow + jb + off);
        acc = __builtin_amdgcn_wmma_f32_16x16x4_f32(
            /*neg_a=*/false, a, /*neg_b=*/false, b,
            /*c_mod=*/(short)0, acc, /*reuse_a=*/false, /*reuse_b=*/false);
    }

    // Epilogue: sinr = y_ii / (rowsum - y_ii + 1)
    const int i = ib + row;                 // column index for this lane
    const float gii = pathloss[i * NL + i];
    #pragma unroll
    for (int r = 0; r < 8; ++r) {
        const int n = nb + half * 8 + r;    // D-matrix row held in acc[r]
        const float yii = gii * powers[n * NL + i];
        const float s = yii / (acc[r] - yii + 1.0f);
        sinr[n * NL + i] = s;
    }
}

// ---------------------------------------------------------------------------
// Kernel B: Q1[l,t] = 1 - exp(-(2^t - 1) * sinr[0,l]);  outa[0,l] = Q1[l,T-1]
// ---------------------------------------------------------------------------
__global__ __launch_bounds__(256) void init_q1_kernel(
    const float* __restrict__ sinr,   // [NB, NL] (row 0 used)
    float* __restrict__ Q1,           // [NL, TT]
    float* __restrict__ outa)         // [NB, NL] (row 0 written)
{
    const int idx = blockIdx.x * 256 + threadIdx.x;
    if (idx >= NL * TT) return;
    const int l = idx / TT;
    const int t = idx - l * TT;
    const float s  = sinr[l];
    const float p2 = exp2f((float)t * DTF);
    const float v  = 1.0f - __expf(-(p2 - 1.0f) * s);
    Q1[idx] = v;
    if (t == TT - 1) outa[l] = v;
}

// ---------------------------------------------------------------------------
// Kernel C: one causal depthwise-conv scan step (one block per link).
//   Qn[l,o] = DT * sum_{j=0..o} Qp[l,j] * qr[l,o-j]
//   qr[l,m] = exp(-(2^k-1) s) * 2^k * s * ln2, k = T-1-m
// Qp and qr staged in LDS (8 KB). Each thread produces 4 outputs.
// ---------------------------------------------------------------------------
__global__ __launch_bounds__(256) void conv_step_kernel(
    const float* __restrict__ Qprev,  // [NL, TT]
    float* __restrict__ Qnext,        // [NL, TT]
    const float* __restrict__ sinr_n, // [NL] (row n of sinr)
    float* __restrict__ outa_n)       // [NL] (row n of outa)
{
    __shared__ float Qp[TT];
    __shared__ float qr[TT];

    const int l   = blockIdx.x;
    const int tid = threadIdx.x;
    const float s = sinr_n[l];

    for (int t = tid; t < TT; t += 256) {
        Qp[t] = Qprev[l * TT + t];
        const int k = (TT - 1) - t;                    // time-reversed kernel
        const float p2 = exp2f((float)k * DTF);
        qr[t] = __expf(-(p2 - 1.0f) * s) * p2 * s * LN2F;
    }
    __syncthreads();

    #pragma unroll
    for (int kk = 0; kk < 4; ++kk) {
        const int o = tid + 256 * kk;
        if (o < TT) {
            float acc = 0.0f;
            #pragma unroll 4
            for (int j = 0; j <= o; ++j)
                acc = fmaf(Qp[j], qr[o - j], acc);     // Qp[j] broadcast, qr consecutive
            const float v = acc * DTF;
            Qnext[l * TT + o] = v;
            if (o == TT - 1) outa_n[l] = v;
        }
    }
}

// ---------------------------------------------------------------------------
// Kernel D: scalar loss reduction.
// loss = sum_l P[0,l]
//      + sum_{m=0..62} sum_l outa[m,l]*(P[m+1,l] + RHO)      (E interf. + D var part)
//      + LAM * sum_l outa[63,l]
//      + RHO * NL                                            (D constant part)
// ---------------------------------------------------------------------------
__global__ __launch_bounds__(256) void reduce_loss_kernel(
    const float* __restrict__ powers, // [NB, NL]
    const float* __restrict__ outa,   // [NB, NL]
    float* __restrict__ out)          // [1]
{
    __shared__ float red[256];
    const int tid = threadIdx.x;
    float acc = 0.0f;

    for (int l = tid; l < NL; l += 256) acc += powers[l];              // P[0,:]
    for (int m = 0; m < NB - 1; ++m)
        for (int l = tid; l < NL; l += 256)
            acc += outa[m * NL + l] * (powers[(m + 1) * NL + l] + 1.0f);
    for (int l = tid; l < NL; l += 256) acc += outa[(NB - 1) * NL + l];

    red[tid] = acc;
    __syncthreads();
    for (int s = 128; s > 0; s >>= 1) {
        if (tid < s) red[tid] += red[tid + s];
        __syncthreads();
    }
    if (tid == 0) out[0] = red[0] + (float)NL;   // + RHO * sum(1)
}

// ---------------------------------------------------------------------------
extern "C" void kernel_launch(void* const* d_in, const int* in_sizes, int n_in,
                              void* d_out, int out_size, void* d_ws, size_t ws_size,
                              hipStream_t stream) {
    (void)in_sizes; (void)n_in; (void)out_size; (void)ws_size;
    const float* powers   = (const float*)d_in[0];  // [64, 1024]
    const float* pathloss = (const float*)d_in[1];  // [1024, 1024]
    float* out = (float*)d_out;                     // scalar loss

    // Workspace layout (floats): sinr | outa | Qa | Qb   (~8.7 MB total)
    float* ws   = (float*)d_ws;
    float* sinr = ws;                         // 64*1024
    float* outa = sinr + NB * NL;             // 64*1024
    float* Qa   = outa + NB * NL;             // 1024*1000
    float* Qb   = Qa + NL * TT;               // 1024*1000

    // Phase 1: SINR via WMMA GEMM (256 waves = 32 blocks x 8 waves)
    sinr_wmma_kernel<<<32, 256, 0, stream>>>(powers, pathloss, sinr);

    // Phase 2: Q1 init + outa[0]
    init_q1_kernel<<<(NL * TT + 255) / 256, 256, 0, stream>>>(sinr, Qa, outa);

    // Phase 3: 63 sequential depthwise-conv steps (ping-pong Q buffers)
    float* Qsrc = Qa;
    float* Qdst = Qb;
    for (int n = 1; n < NB; ++n) {
        conv_step_kernel<<<NL, 256, 0, stream>>>(Qsrc, Qdst, sinr + n * NL, outa + n * NL);
        float* tmp = Qsrc; Qsrc = Qdst; Qdst = tmp;
    }

    // Phase 4: scalar loss
    reduce_loss_kernel<<<1, 256, 0, stream>>>(powers, outa, out);
}